// _ModewisePredictionProjector_89060441850391
// MI455X (gfx1250) — compile-verified
//
#include <hip/hip_runtime.h>

typedef __attribute__((ext_vector_type(2))) float v2f;
typedef __attribute__((ext_vector_type(4))) float v4f;
typedef __attribute__((ext_vector_type(8))) float v8f;

#define TILE 4096  // 16*16*16 per (b,s)

__device__ __forceinline__ v8f wmma4(v2f a, v2f b, v8f c) {
  // D(16x16,f32) = A(16x4,f32) x B(4x16,f32) + C
  return __builtin_amdgcn_wmma_f32_16x16x4_f32(
      /*neg_a=*/false, a, /*neg_b=*/false, b,
      /*c_mod=*/(short)0, c, /*reuse_a=*/false, /*reuse_b=*/false);
}

__global__ __launch_bounds__(128)
void modewise_proj_kernel(const float* __restrict__ xr, const float* __restrict__ xi,
                          const float* __restrict__ wr0, const float* __restrict__ wi0,
                          const float* __restrict__ wr1, const float* __restrict__ wi1,
                          const int* __restrict__ p_active0, const int* __restrict__ p_active1,
                          const int* __restrict__ p_state_rank, const int* __restrict__ p_active_rank,
                          float* __restrict__ outr, float* __restrict__ outi) {
  __shared__ float Xr[TILE];
  __shared__ float Xi[TILE];

  const int tid  = threadIdx.x;
  const int lane = tid & 31;
  const int wave = tid >> 5;
  const int half = lane >> 4;   // which 16-lane half
  const int ln   = lane & 15;
  const long long base = (long long)blockIdx.x * TILE;

  // ---- cooperative coalesced load of the 16x16x16 complex tile into LDS
  {
    const v4f* gxr = (const v4f*)(xr + base);
    const v4f* gxi = (const v4f*)(xi + base);
    v4f* sxr = (v4f*)Xr;
    v4f* sxi = (v4f*)Xi;
#pragma unroll
    for (int i = 0; i < 8; ++i) {
      int idx = tid + i * 128;
      sxr[idx] = gxr[idx];
      sxi[idx] = gxi[idx];
    }
  }

  const int active0     = p_active0[0];
  const int active1     = p_active1[0];
  const int state_rank  = p_state_rank[0];
  const int active_rank = p_active_rank[0];

  // ---- build A-matrix fragments for the two coupled modes (identity if inactive)
  // A 16x4 f32 layout: lane holds M = lane&15 ; element v holds K = kc*4 + 2*(lane>>4) + v
  v2f Ar[2][4], Ai[2][4], Ain[2][4];
#pragma unroll
  for (int m = 0; m < 2; ++m) {
    const float* wr = (m == 0) ? wr0 : wr1;
    const float* wi = (m == 0) ? wi0 : wi1;
    const int act = (m == 0) ? active0 : active1;
    const bool coupled = (m < state_rank) && (m < active_rank);
#pragma unroll
    for (int kc = 0; kc < 4; ++kc) {
#pragma unroll
      for (int v = 0; v < 2; ++v) {
        const int M = ln;
        const int K = kc * 4 + 2 * half + v;
        float ar, ai;
        if (coupled) {
          const bool in = (M < act) && (K < act);  // zero-padded coupling block
          ar = in ? wr[M * 16 + K] : 0.0f;
          ai = in ? wi[M * 16 + K] : 0.0f;
        } else {
          ar = (M == K) ? 1.0f : 0.0f;             // identity bypass
          ai = 0.0f;
        }
        Ar[m][kc][v]  = ar;
        Ai[m][kc][v]  = ai;
        Ain[m][kc][v] = -ai;                       // for real part: Ar*Br - Ai*Bi
      }
    }
  }

  __syncthreads();

  // ---- Mode 0: y[i, n] = sum_j C0[i,j] * x[j, n], n = m1*16+m2 in [0,256)
  // wave owns column blocks cb = wave*4 .. wave*4+3 -> in-place update is safe
#pragma unroll
  for (int q = 0; q < 4; ++q) {
    const int cb  = wave * 4 + q;
    const int col = cb * 16 + ln;
    v8f Dr = {}; v8f Di = {};
#pragma unroll
    for (int kc = 0; kc < 4; ++kc) {
      const int j0 = kc * 4 + 2 * half;  // B layout: element v -> K = j0+v, N = ln
      v2f Br, Bi;
      Br[0] = Xr[(j0 + 0) * 256 + col];
      Br[1] = Xr[(j0 + 1) * 256 + col];
      Bi[0] = Xi[(j0 + 0) * 256 + col];
      Bi[1] = Xi[(j0 + 1) * 256 + col];
      Dr = wmma4(Ar[0][kc],  Br, Dr);
      Di = wmma4(Ai[0][kc],  Br, Di);
      Dr = wmma4(Ain[0][kc], Bi, Dr);
      Di = wmma4(Ar[0][kc],  Bi, Di);
    }
#pragma unroll
    for (int r = 0; r < 8; ++r) {
      const int i = r + 8 * half;        // D layout: element r -> row i, col N
      Xr[i * 256 + col] = Dr[r];
      Xi[i * 256 + col] = Di[r];
    }
  }

  __syncthreads();

  // ---- Mode 1: y[m0, i, m2] = sum_j C1[i,j] * x[m0, j, m2]
  // wave owns m0 = wave*4 .. wave*4+3 -> in-place update is safe
#pragma unroll
  for (int q = 0; q < 4; ++q) {
    const int m0 = wave * 4 + q;
    const int rowbase = m0 * 256;
    v8f Dr = {}; v8f Di = {};
#pragma unroll
    for (int kc = 0; kc < 4; ++kc) {
      const int j0 = kc * 4 + 2 * half;
      v2f Br, Bi;
      Br[0] = Xr[rowbase + (j0 + 0) * 16 + ln];
      Br[1] = Xr[rowbase + (j0 + 1) * 16 + ln];
      Bi[0] = Xi[rowbase + (j0 + 0) * 16 + ln];
      Bi[1] = Xi[rowbase + (j0 + 1) * 16 + ln];
      Dr = wmma4(Ar[1][kc],  Br, Dr);
      Di = wmma4(Ai[1][kc],  Br, Di);
      Dr = wmma4(Ain[1][kc], Bi, Dr);
      Di = wmma4(Ar[1][kc],  Bi, Di);
    }
#pragma unroll
    for (int r = 0; r < 8; ++r) {
      const int i = r + 8 * half;
      Xr[rowbase + i * 16 + ln] = Dr[r];
      Xi[rowbase + i * 16 + ln] = Di[r];
    }
  }

  __syncthreads();

  // ---- Mode 2: k=2 >= active_rank for these inputs -> identity bypass (no-op)

  // ---- coalesced store
  {
    v4f* goutr = (v4f*)(outr + base);
    v4f* gouti = (v4f*)(outi + base);
    const v4f* sxr = (const v4f*)Xr;
    const v4f* sxi = (const v4f*)Xi;
#pragma unroll
    for (int i = 0; i < 8; ++i) {
      int idx = tid + i * 128;
      goutr[idx] = sxr[idx];
      gouti[idx] = sxi[idx];
    }
  }
}

extern "C" void kernel_launch(void* const* d_in, const int* in_sizes, int n_in,
                              void* d_out, int out_size, void* d_ws, size_t ws_size,
                              hipStream_t stream) {
  // input order: real, imag, w_real_0, w_imag_0, w_real_1, w_imag_1,
  //              w_real_2, w_imag_2, active0, active1, active2, state_rank, active_rank
  const float* xr  = (const float*)d_in[0];
  const float* xi  = (const float*)d_in[1];
  const float* wr0 = (const float*)d_in[2];
  const float* wi0 = (const float*)d_in[3];
  const float* wr1 = (const float*)d_in[4];
  const float* wi1 = (const float*)d_in[5];
  // d_in[6]/d_in[7] (w_real_2/w_imag_2) unused: mode 2 is identity for active_rank<=2
  const int* active0     = (const int*)d_in[8];
  const int* active1     = (const int*)d_in[9];
  const int* state_rank  = (const int*)d_in[11];
  const int* active_rank = (const int*)d_in[12];

  const int n      = in_sizes[0];   // B*S*16*16*16
  const int ntiles = n / TILE;      // B*S = 8192
  float* outr = (float*)d_out;      // tuple output: real then imag, concatenated
  float* outi = (float*)d_out + n;

  hipLaunchKernelGGL(modewise_proj_kernel, dim3(ntiles), dim3(128), 0, stream,
                     xr, xi, wr0, wi0, wr1, wi1,
                     active0, active1, state_rank, active_rank, outr, outi);
}